// VQ_VAE_26156350832920
// MI455X (gfx1250) — compile-verified
//
#include <hip/hip_runtime.h>
#include <math.h>

typedef __attribute__((ext_vector_type(2))) float v2f;
typedef __attribute__((ext_vector_type(4))) float v4f;
typedef __attribute__((ext_vector_type(8))) float v8f;

#define D_DIM   256
#define K_CODES 2048
#define N_ROWS  65536
#define HW_SZ   4096      // 64*64
#define XP      258       // padded LDS row pitch (258 % 64 = 2 -> conflict-free b64 loads)

// ---- workspace layout (float offsets) ----
#define WS_WW     0                      // 2048   : ||w_k||^2
#define WS_COUNTS 2048                   // 2048   : histogram
#define WS_DW     4096                   // 524288 : encodings^T @ flat
#define WS_CSPRE  528384                 // 2048   : pre-normalized cluster size
#define WS_ACC    530432                 // 3      : [loss, n_sum, entropy]
#define WS_IDX    530436                 // 65536 ints

// ---- output layout (float offsets, reference tuple order, flattened) ----
#define O_LOSS  ((size_t)0)
#define O_QOUT  ((size_t)1)
#define O_PERP  ((size_t)16777217)
#define O_ENC   ((size_t)16777218)
#define O_CS    ((size_t)151994946)
#define O_EMAW  ((size_t)151996994)
#define O_NW    ((size_t)152521282)

// ||weight[k]||^2, one thread per code
__global__ void ww_kernel(const float* __restrict__ w, float* __restrict__ ww) {
    int k = blockIdx.x * 256 + threadIdx.x;
    const v4f* wr = (const v4f*)(w + (size_t)k * D_DIM);
    float s = 0.f;
    #pragma unroll 8
    for (int i = 0; i < D_DIM / 4; ++i) {
        v4f v = wr[i];
        s += v.x * v.x + v.y * v.y + v.z * v.z + v.w * v.w;
    }
    ww[k] = s;
}

// zero counts + dw + accumulators (ws is poisoned with 0xAA by the harness)
__global__ void zero_kernel(float* __restrict__ ws) {
    int gid = blockIdx.x * 256 + threadIdx.x;
    if (gid < (WS_CSPRE - WS_COUNTS)) ws[WS_COUNTS + gid] = 0.f;  // counts + dw (contiguous)
    if (gid < 3) ws[WS_ACC + gid] = 0.f;
}

// Distance GEMM (f32 WMMA) + per-row argmin over the codebook.
// Block = 128 threads = 4 waves; each wave owns a 16-row x tile in LDS
// (the compiler hoists the A operands into VGPRs after the transpose-stage).
__global__ __launch_bounds__(128) void vq_argmin_wmma(
    const float* __restrict__ z, const float* __restrict__ weight,
    const float* __restrict__ ww, int* __restrict__ idx_out)
{
    extern __shared__ float xs[];                       // 4 * 16 * XP floats
    const int tid  = threadIdx.x;
    const int wave = tid >> 5;
    const int lane = tid & 31;
    const int m    = lane & 15;                         // row (A/C) or column (B/C) slot
    const int hi   = lane >> 4;                         // half-wave select
    const int row0 = blockIdx.x * 64 + wave * 16;       // first row of this wave's tile
    const int b    = row0 >> 12;                        // batch  (HW = 4096)
    const int hw0  = row0 & (HW_SZ - 1);

    // Stage x tile [16 rows x 256] into LDS, transposing from z's [B, D, H, W] layout.
    // Rows are 16-aligned so all 16 share (b) and hw0..hw0+15 is contiguous.
    float* xw = xs + wave * 16 * XP;
    const size_t zb = (size_t)b * (D_DIM * HW_SZ) + hw0 + m;
    for (int db = 0; db < D_DIM; db += 2) {
        int d = db + hi;
        xw[m * XP + d] = z[zb + (size_t)d * HW_SZ];
    }
    __syncthreads();

    float rmin[8];
    int   ridx[8];
    #pragma unroll
    for (int r = 0; r < 8; ++r) { rmin[r] = 3.4e38f; ridx[r] = 0; }

    const float* xrow = xw + m * XP;                    // A: lane's matrix row
    for (int t = 0; t < 128; ++t) {
        const int col = t * 16 + m;                     // B/C: lane's matrix column
        const float* wrow = weight + (size_t)col * D_DIM;
        if (t < 127) __builtin_prefetch(wrow + 16 * D_DIM, 0, 1);  // next tile
        const float wwv = ww[col];

        v8f c = {};
        #pragma unroll
        for (int kb = 0; kb < D_DIM; kb += 4) {
            v2f a = *(const v2f*)(xrow + kb + 2 * hi);  // A 16x4: K = 2*hi, 2*hi+1
            v2f bb = *(const v2f*)(wrow + kb + 2 * hi); // B 4x16: B[k][n] = W[n][k]
            c = __builtin_amdgcn_wmma_f32_16x16x4_f32(
                    false, a, false, bb, (short)0, c, false, false);
        }
        // C layout: VGPR r holds row (r + 8*hi), column = m, for this lane.
        #pragma unroll
        for (int r = 0; r < 8; ++r) {
            float dist = wwv - 2.0f * c[r];             // ||x||^2 dropped (row-constant)
            if (dist < rmin[r]) { rmin[r] = dist; ridx[r] = col; }
        }
    }

    // 16-lane butterfly min+argmin (halves are independent rows 0-7 / 8-15).
    #pragma unroll
    for (int r = 0; r < 8; ++r) {
        float mv = rmin[r]; int mi = ridx[r];
        #pragma unroll
        for (int s = 1; s < 16; s <<= 1) {
            float ov = __shfl_xor(mv, s, 32);
            int   oi = __shfl_xor(mi, s, 32);
            if (ov < mv || (ov == mv && oi < mi)) { mv = ov; mi = oi; }  // first-min tie-break
        }
        rmin[r] = mv; ridx[r] = mi;
    }
    if (m == 0) {
        int rb = row0 + hi * 8;
        #pragma unroll
        for (int r = 0; r < 8; ++r) idx_out[rb + r] = ridx[r];
    }
}

// one-hot encodings: 512 MB write-once stream = the bandwidth floor.
// b128 non-temporal stores: 4x fewer VMEM instructions, no L2 pollution.
__global__ void enc_kernel(const int* __restrict__ idx, float* __restrict__ enc,
                           float* __restrict__ counts) {
    const int row = blockIdx.x, t = threadIdx.x;
    const int id = idx[row];
    v4f* e = (v4f*)(enc + (size_t)row * K_CODES);
    #pragma unroll
    for (int j = 0; j < K_CODES / (256 * 4); ++j) {
        const int v = j * 256 + t;                      // float4 index; cols 4v..4v+3
        const int c0 = v * 4;
        v4f ov = { c0 == id ? 1.f : 0.f, c0 + 1 == id ? 1.f : 0.f,
                   c0 + 2 == id ? 1.f : 0.f, c0 + 3 == id ? 1.f : 0.f };
        __builtin_nontemporal_store(ov, e + v);
    }
    if (t == 0) atomicAdd(&counts[id], 1.0f);
}

// quantized output (numerically == straight-through), dw scatter-add, commitment loss
__global__ void quant_kernel(const float* __restrict__ z, const float* __restrict__ w,
                             const int* __restrict__ idx, float* __restrict__ qout,
                             float* __restrict__ dw, float* __restrict__ loss_acc) {
    const size_t gid = (size_t)blockIdx.x * 256 + threadIdx.x;   // over 16 * 2^20
    const int bb = (int)(gid >> 20);
    const int rem = (int)(gid & ((1u << 20) - 1));
    const int d  = rem >> 12;
    const int hw = rem & (HW_SZ - 1);
    const int id = idx[(bb << 12) | hw];
    const float wv = w[(size_t)id * D_DIM + d];
    const float zv = z[gid];
    __builtin_nontemporal_store(wv, qout + gid);      // x + sg(q - x) == q numerically
    atomicAdd(&dw[(size_t)id * D_DIM + d], zv);       // encodings^T @ flat
    float df = (wv - zv) * (wv - zv);
    #pragma unroll
    for (int s = 16; s > 0; s >>= 1) df += __shfl_xor(df, s, 32);
    if ((threadIdx.x & 31) == 0) atomicAdd(loss_acc, df);
}

// EMA cluster size (pre-normalization), n = sum(cs), entropy for perplexity
__global__ void cs_kernel(const float* __restrict__ counts, const float* __restrict__ ema_cs,
                          float* __restrict__ cs_pre, float* __restrict__ acc) {
    const int k = blockIdx.x * 256 + threadIdx.x;
    const float cnt = counts[k];
    float cp = ema_cs[k] * 0.99f + 0.01f * cnt;
    cs_pre[k] = cp;
    float p = cnt * (1.0f / N_ROWS);
    float ent = -p * logf(p + 1e-10f);
    #pragma unroll
    for (int s = 16; s > 0; s >>= 1) {
        cp  += __shfl_xor(cp, s, 32);
        ent += __shfl_xor(ent, s, 32);
    }
    if ((threadIdx.x & 31) == 0) { atomicAdd(&acc[1], cp); atomicAdd(&acc[2], ent); }
}

// cs normalization, new_ema_w, new_weight, loss & perplexity scalars
__global__ void final_kernel(const float* __restrict__ ema_w, const float* __restrict__ dw,
                             const float* __restrict__ cs_pre, const float* __restrict__ acc,
                             float* __restrict__ loss_out, float* __restrict__ perp_out,
                             float* __restrict__ cs_out, float* __restrict__ emaw_out,
                             float* __restrict__ nw_out) {
    const size_t gid = (size_t)blockIdx.x * 256 + threadIdx.x;   // over K*D = 524288
    const int k = (int)(gid >> 8), d = (int)(gid & 255);
    const float nS = acc[1];
    const float csf = (cs_pre[k] + 1e-5f) / (nS + K_CODES * 1e-5f) * nS;
    const float ew = ema_w[gid] * 0.99f + 0.01f * dw[gid];
    emaw_out[gid] = ew;
    nw_out[gid] = ew / csf;
    if (d == 0) cs_out[k] = csf;
    if (gid == 0) {
        *loss_out = 0.25f * acc[0] * (1.0f / 16777216.0f);
        *perp_out = expf(acc[2]);
    }
}

extern "C" void kernel_launch(void* const* d_in, const int* in_sizes, int n_in,
                              void* d_out, int out_size, void* d_ws, size_t ws_size,
                              hipStream_t stream) {
    const float* z      = (const float*)d_in[0];
    const float* weight = (const float*)d_in[1];
    const float* ema_cs = (const float*)d_in[2];
    const float* ema_w  = (const float*)d_in[3];
    float* out = (float*)d_out;
    float* ws  = (float*)d_ws;
    int*   idx = (int*)(ws + WS_IDX);

    ww_kernel  <<<K_CODES / 256, 256, 0, stream>>>(weight, ws + WS_WW);
    zero_kernel<<<(WS_CSPRE - WS_COUNTS + 255) / 256, 256, 0, stream>>>(ws);
    vq_argmin_wmma<<<N_ROWS / 64, 128, 4 * 16 * XP * sizeof(float), stream>>>(
        z, weight, ws + WS_WW, idx);
    enc_kernel <<<N_ROWS, 256, 0, stream>>>(idx, out + O_ENC, ws + WS_COUNTS);
    quant_kernel<<<16777216 / 256, 256, 0, stream>>>(
        z, weight, idx, out + O_QOUT, ws + WS_DW, ws + WS_ACC + 0);
    cs_kernel  <<<K_CODES / 256, 256, 0, stream>>>(ws + WS_COUNTS, ema_cs,
                                                   ws + WS_CSPRE, ws + WS_ACC);
    final_kernel<<<(K_CODES * D_DIM) / 256, 256, 0, stream>>>(
        ema_w, ws + WS_DW, ws + WS_CSPRE, ws + WS_ACC,
        out + O_LOSS, out + O_PERP, out + O_CS, out + O_EMAW, out + O_NW);
}